// NegSimHead_9672266351058
// MI455X (gfx1250) — compile-verified
//
#include <hip/hip_runtime.h>
#include <hip/hip_bf16.h>
#include <math.h>

// N=8192 rows, C=512 features, fp32 everywhere (matches reference precision).
// Score GEMMs run on the fp32 matrix pipe: V_WMMA_F32_16X16X4_F32.
//
// Normalized matrices are stored in WMMA-FRAGMENT ORDER (valid because dot
// products are invariant under a feature permutation applied to both operands,
// and A/B fragment layouts coincide for X @ Y^T):
//   swz[g][sp][lane][0..3], g = 16-row group, sp = k-step pair (0..63)
//   lane -> (m = lane&15, kh = lane>>4); floats f0..f3 = features
//   {8sp+2kh, 8sp+2kh+1, 8sp+4+2kh, 8sp+5+2kh} of row 16g+m.
// => every GEMM operand load is ONE coalesced 512B global_load_b128 with
//    immediate offsets off a single base pointer.

#define NN 8192
#define CC 512

typedef float v2f __attribute__((ext_vector_type(2)));
typedef float v4f __attribute__((ext_vector_type(4)));
typedef float v8f __attribute__((ext_vector_type(8)));

// ---------------- zero-init integer rank counters ----------------
__global__ void k_zero(int* __restrict__ p, int n) {
    int i = blockIdx.x * blockDim.x + threadIdx.x;
    if (i < n) p[i] = 0;
}

// ---------------- per-row inverse L2 norms (coalesced) ----------------
// grid (8192 rows, 4 matrices), block 128
__global__ void __launch_bounds__(128) k_norms(
    const float* __restrict__ i0, const float* __restrict__ i1,
    const float* __restrict__ i2, const float* __restrict__ i3,
    float* __restrict__ invn)
{
    const float* src = (blockIdx.y == 0) ? i0 : (blockIdx.y == 1) ? i1
                     : (blockIdx.y == 2) ? i2 : i3;
    int t = threadIdx.x;
    const float4* s4 = (const float4*)(src + (size_t)blockIdx.x * CC);
    float4 v = s4[t];
    float ss = v.x * v.x + v.y * v.y + v.z * v.z + v.w * v.w;
    for (int o = 16; o > 0; o >>= 1) ss += __shfl_down(ss, o);
    __shared__ float ls[4];
    if ((t & 31) == 0) ls[t >> 5] = ss;
    __syncthreads();
    if (t == 0) {
        float tot = ls[0] + ls[1] + ls[2] + ls[3];
        invn[(size_t)blockIdx.y * NN + blockIdx.x] = 1.0f / sqrtf(tot);
    }
}

// ------------- swizzle: raw * inv_norm -> fragment-ordered layout -------------
// grid (512 groups, 4 matrices), block 256; output stores fully coalesced.
__global__ void __launch_bounds__(256) k_swizzle(
    const float* __restrict__ i0, const float* __restrict__ i1,
    const float* __restrict__ i2, const float* __restrict__ i3,
    const float* __restrict__ invn, float* __restrict__ Wns)
{
    const float* src = (blockIdx.y == 0) ? i0 : (blockIdx.y == 1) ? i1
                     : (blockIdx.y == 2) ? i2 : i3;
    int g = blockIdx.x;
    float* dst = Wns + (size_t)blockIdx.y * NN * CC + (size_t)g * (64 * 32 * 4);
    const float* inv_base = invn + (size_t)blockIdx.y * NN;
#pragma unroll
    for (int pass = 0; pass < 8; ++pass) {
        int so = pass * 256 + threadIdx.x;      // output b128 slot (0..2047)
        int sp = so >> 5;
        int lane_o = so & 31;
        int m = lane_o & 15, kh = lane_o >> 4;
        int row = g * 16 + m;
        float inv = inv_base[row];
        const float* rp = src + (size_t)row * CC + 8 * sp + 2 * kh;
        float2 x01 = *(const float2*)(rp);      // features 8sp+2kh, +1
        float2 x23 = *(const float2*)(rp + 4);  // features 8sp+4+2kh, +1
        v4f o;
        o[0] = x01.x * inv; o[1] = x01.y * inv;
        o[2] = x23.x * inv; o[3] = x23.y * inv;
        *(v4f*)(dst + (size_t)so * 4) = o;
    }
}

// ------------- per-column sum / sumsq partials (raw * inv_norm) -------------
// Exactly the reference's columns (no permutation involved).
// grid: (16 colgroups, 8 rowchunks, 4 matrices), block 256
__global__ void __launch_bounds__(256) k_colpart(
    const float* __restrict__ i0, const float* __restrict__ i1,
    const float* __restrict__ i2, const float* __restrict__ i3,
    const float* __restrict__ invn,
    float* __restrict__ psum, float* __restrict__ psq)
{
    int mtx = blockIdx.z, cg = blockIdx.x, rc = blockIdx.y;
    const float* base = (mtx == 0) ? i0 : (mtx == 1) ? i1 : (mtx == 2) ? i2 : i3;
    const float* inv_base = invn + (size_t)mtx * NN;
    int t = threadIdx.x;
    int cl = t & 31;           // column within group (32 cols -> coalesced 128B)
    int stripe = t >> 5;       // 8 row stripes
    int c = cg * 32 + cl;
    float s = 0.f, q = 0.f;
    int r0 = rc * 1024 + stripe * 128;
    for (int it = 0; it < 128; ++it) {
        int r = r0 + it;
        float x = base[(size_t)r * CC + c] * inv_base[r];
        s += x; q += x * x;
    }
    __shared__ float ls[256], lq[256];
    ls[t] = s; lq[t] = q;
    __syncthreads();
    if (stripe == 0) {
        for (int k = 1; k < 8; ++k) { s += ls[cl + k * 32]; q += lq[cl + k * 32]; }
        int idx = ((mtx * 16 + cg) * 8 + rc) * 32 + cl;
        psum[idx] = s; psq[idx] = q;
    }
}

// ------------- diagonal pass: WMMA on diagonal 16x16 tiles -------------
// Identical instruction/feature grouping as the count pass => bitwise-equal
// s[i][i], so the self element ties exactly (stable-argsort semantics).
// grid: (512 tiles, 2 gemms), block 32 (one full wave, EXEC all ones)
__global__ void __launch_bounds__(32) k_diag(
    const float* __restrict__ Wns, float* __restrict__ diag1, float* __restrict__ diag2)
{
    const size_t MSZ = (size_t)NN * CC;
    const float* X; const float* Y; float* dg;
    if (blockIdx.y == 0) { X = Wns;       Y = Wns + 3 * MSZ; dg = diag1; }  // v_n @ p_t_n^T
    else                 { X = Wns + MSZ; Y = Wns + 2 * MSZ; dg = diag2; }  // t_n @ p_v_n^T
    int g = blockIdx.x;
    int lane = threadIdx.x;
    const float* abase = X + (size_t)g * 8192 + lane * 4;   // 8192 floats per group
    const float* bbase = Y + (size_t)g * 8192 + lane * 4;
    v8f c = {};
#pragma unroll 4
    for (int sp = 0; sp < 64; ++sp) {          // 2 k-steps per coalesced b128
        v4f a = *(const v4f*)(abase + sp * 128);
        v4f b = *(const v4f*)(bbase + sp * 128);
        v2f alo = __builtin_shufflevector(a, a, 0, 1);
        v2f ahi = __builtin_shufflevector(a, a, 2, 3);
        v2f blo = __builtin_shufflevector(b, b, 0, 1);
        v2f bhi = __builtin_shufflevector(b, b, 2, 3);
        c = __builtin_amdgcn_wmma_f32_16x16x4_f32(false, alo, false, blo, (short)0, c, false, false);
        c = __builtin_amdgcn_wmma_f32_16x16x4_f32(false, ahi, false, bhi, (short)0, c, false, false);
    }
    int m = lane & 15, kh = lane >> 4;
    int r = m - kh * 8;
    if (r >= 0 && r < 8) dg[g * 16 + m] = c[r];
}

// ------------- count pass: full GEMM tiles -> per-row rank counts -------------
// grid: (64 col-blocks of 128, 64 row-blocks of 128, 2 gemms), block 256 (8 waves)
// wave w: rows i0 = by*128 + 16w, cols j0..j0+127 (8 tiles reuse one A fragment;
// 9 coalesced 512B loads per 16 WMMAs, all immediate-offset addressed)
__global__ void __launch_bounds__(256) k_count(
    const float* __restrict__ Wns, const float* __restrict__ diag1,
    const float* __restrict__ diag2, int* __restrict__ pos1, int* __restrict__ pos2)
{
    const size_t MSZ = (size_t)NN * CC;
    const float* X; const float* Y; const float* dg; int* pos;
    if (blockIdx.z == 0) { X = Wns;       Y = Wns + 3 * MSZ; dg = diag1; pos = pos1; }
    else                 { X = Wns + MSZ; Y = Wns + 2 * MSZ; dg = diag2; pos = pos2; }

    int w = threadIdx.x >> 5;
    int lane = threadIdx.x & 31;
    int gi = blockIdx.y * 8 + w;               // row group of this wave
    int gj = blockIdx.x * 8;                   // first col group of the block
    int i0 = gi * 16;
    int j0 = gj * 16;

    const float* abase = X + (size_t)gi * 8192 + lane * 4;
    const float* bbase = Y + (size_t)gj * 8192 + lane * 4;

    v8f acc[8] = {};
    for (int sp = 0; sp < 64; ++sp) {          // 2 k-steps per coalesced b128
        v4f a = *(const v4f*)(abase + sp * 128);
        v2f alo = __builtin_shufflevector(a, a, 0, 1);
        v2f ahi = __builtin_shufflevector(a, a, 2, 3);
#pragma unroll
        for (int t8 = 0; t8 < 8; ++t8) {
            v4f b = *(const v4f*)(bbase + t8 * 8192 + sp * 128);
            v2f blo = __builtin_shufflevector(b, b, 0, 1);
            v2f bhi = __builtin_shufflevector(b, b, 2, 3);
            acc[t8] = __builtin_amdgcn_wmma_f32_16x16x4_f32(
                false, alo, false, blo, (short)0, acc[t8], false, false);
            acc[t8] = __builtin_amdgcn_wmma_f32_16x16x4_f32(
                false, ahi, false, bhi, (short)0, acc[t8], false, false);
        }
    }

    // rank counting: pos_i += #{s_ij > s_ii} + #{s_ij == s_ii && j < i}
    __shared__ int cnt[128];
    if (threadIdx.x < 128) cnt[threadIdx.x] = 0;
    __syncthreads();

    int m = lane & 15, kh = lane >> 4;
    int mb = kh * 8;
    int rcnt[8] = {};
#pragma unroll
    for (int t8 = 0; t8 < 8; ++t8) {
        int col = j0 + t8 * 16 + m;
#pragma unroll
        for (int r = 0; r < 8; ++r) {
            int row = i0 + mb + r;
            float val = acc[t8][r];
            float d = dg[row];
            rcnt[r] += ((val > d) || (val == d && col < row)) ? 1 : 0;
        }
    }
#pragma unroll
    for (int r = 0; r < 8; ++r) atomicAdd(&cnt[w * 16 + mb + r], rcnt[r]);
    __syncthreads();
    if (threadIdx.x < 128)
        atomicAdd(&pos[blockIdx.y * 128 + threadIdx.x], cnt[threadIdx.x]);  // integer: deterministic
}

// ---------------- final reduction: 13 outputs ----------------
__device__ float blockReduceF(float v, float* red) {
    int t = threadIdx.x;
    red[t] = v; __syncthreads();
    for (int o = 128; o > 0; o >>= 1) { if (t < o) red[t] += red[t + o]; __syncthreads(); }
    float r = red[0]; __syncthreads();
    return r;
}
__device__ int blockReduceI(int v, int* red) {
    int t = threadIdx.x;
    red[t] = v; __syncthreads();
    for (int o = 128; o > 0; o >>= 1) { if (t < o) red[t] += red[t + o]; __syncthreads(); }
    int r = red[0]; __syncthreads();
    return r;
}

__global__ void __launch_bounds__(256) k_final(
    const float* __restrict__ diag1, const float* __restrict__ diag2,
    const int* __restrict__ pos1, const int* __restrict__ pos2,
    const float* __restrict__ psum, const float* __restrict__ psq,
    float* __restrict__ out)
{
    __shared__ float redf[256];
    __shared__ int redi[256];
    int t = threadIdx.x;

    float s1 = 0.f, s2 = 0.f;
    int a1 = 0, a5 = 0, a10 = 0, as = 0, b1 = 0, b5 = 0, b10 = 0, bs = 0;
    for (int i = t; i < NN; i += 256) {
        s1 += diag1[i]; s2 += diag2[i];
        int p = pos1[i]; a1 += (p < 1); a5 += (p < 5); a10 += (p < 10); as += p;
        int q = pos2[i]; b1 += (q < 1); b5 += (q < 5); b10 += (q < 10); bs += q;
    }
    float S1 = blockReduceF(s1, redf);
    float S2 = blockReduceF(s2, redf);
    int A1 = blockReduceI(a1, redi), A5 = blockReduceI(a5, redi);
    int A10 = blockReduceI(a10, redi), AS = blockReduceI(as, redi);
    int B1 = blockReduceI(b1, redi), B5 = blockReduceI(b5, redi);
    int B10 = blockReduceI(b10, redi), BS = blockReduceI(bs, redi);

    float st[4];
    for (int mtx = 0; mtx < 4; ++mtx) {
        float acc = 0.f;
        for (int c = t; c < CC; c += 256) {
            float s = 0.f, q = 0.f;
            for (int rc = 0; rc < 8; ++rc) {
                int idx = ((mtx * 16 + (c >> 5)) * 8 + rc) * 32 + (c & 31);
                s += psum[idx]; q += psq[idx];
            }
            float var = (q - s * s / (float)NN) / (float)(NN - 1);  // ddof=1
            if (var < 0.f) var = 0.f;
            acc += sqrtf(var);
        }
        st[mtx] = blockReduceF(acc, redf) / (float)CC;
    }

    if (t == 0) {
        const float invN = 1.0f / (float)NN;
        out[0]  = -0.5f * (S1 + S2) * invN;       // neg_sim_loss
        out[1]  = st[0];  out[2] = st[1];  out[3] = st[2];  out[4] = st[3];
        out[5]  = (float)A1 * invN;  out[6]  = (float)A5 * invN;
        out[7]  = (float)A10 * invN; out[8]  = (float)AS * invN;   // v_mr
        out[9]  = (float)B1 * invN;  out[10] = (float)B5 * invN;
        out[11] = (float)B10 * invN; out[12] = (float)BS * invN;   // t_mr
    }
}

// ---------------- launcher ----------------
extern "C" void kernel_launch(void* const* d_in, const int* in_sizes, int n_in,
                              void* d_out, int out_size, void* d_ws, size_t ws_size,
                              hipStream_t stream) {
    (void)in_sizes; (void)n_in; (void)out_size; (void)ws_size;
    const float* v  = (const float*)d_in[0];
    const float* tf = (const float*)d_in[1];
    const float* pv = (const float*)d_in[2];
    const float* pt = (const float*)d_in[3];
    float* out = (float*)d_out;

    const size_t MSZ = (size_t)NN * CC;
    float* Wns   = (float*)d_ws;          // 4 fragment-ordered matrices: 64 MB (L2-resident)
    float* invn  = Wns + 4 * MSZ;         // 4*8192 row inverse norms
    float* diag1 = invn + 4 * NN;
    float* diag2 = diag1 + NN;
    int*   pos1  = (int*)(diag2 + NN);
    int*   pos2  = pos1 + NN;
    float* psum  = (float*)(pos2 + NN);   // 4*16*8*32 = 16384 floats
    float* psq   = psum + 4 * 16 * 8 * 32;

    k_zero<<<dim3((2 * NN + 255) / 256), 256, 0, stream>>>(pos1, 2 * NN);
    k_norms<<<dim3(NN, 4), 128, 0, stream>>>(v, tf, pv, pt, invn);
    k_swizzle<<<dim3(NN / 16, 4), 256, 0, stream>>>(v, tf, pv, pt, invn, Wns);
    k_colpart<<<dim3(16, 8, 4), 256, 0, stream>>>(v, tf, pv, pt, invn, psum, psq);
    k_diag<<<dim3(NN / 16, 2), 32, 0, stream>>>(Wns, diag1, diag2);
    k_count<<<dim3(NN / 128, NN / 128, 2), 256, 0, stream>>>(Wns, diag1, diag2, pos1, pos2);
    k_final<<<1, 256, 0, stream>>>(diag1, diag2, pos1, pos2, psum, psq, out);
}